// SingleHeadAttention_5257039970418
// MI455X (gfx1250) — compile-verified
//
#include <hip/hip_runtime.h>
#include <math.h>
#include <stdint.h>

// ---------------------------------------------------------------------------
// Single-head causal attention, B=4, T=2048, D=1024, fp32 throughout.
// Stage 1: Q/K/V = X @ W{q,k,v}  (LDS-tiled fp32 WMMA GEMM, async staging)
// Stage 2: flash-attention with streaming softmax (fp32 WMMA for QK^T and PV)
// ---------------------------------------------------------------------------

#define BATCH 4
#define SEQ   2048
#define DIM   1024
#define NROWS (BATCH * SEQ)          // 8192 rows of X/Q/K/V
#define SCALE 0.03125f               // 1/sqrt(1024)

typedef float v2f __attribute__((ext_vector_type(2)));
typedef float v8f __attribute__((ext_vector_type(8)));

__device__ __forceinline__ v8f wmma_f32(v2f a, v2f b, v8f c) {
  // V_WMMA_F32_16X16X4_F32 : D(16x16 f32) = A(16x4 f32) * B(4x16 f32) + C
  return __builtin_amdgcn_wmma_f32_16x16x4_f32(false, a, false, b,
                                               (short)0, c, false, false);
}

// CDNA5 async global->LDS DMA (tracked by ASYNCcnt). VDST operand carries the
// per-lane LDS byte address; low 32 bits of a generic __shared__ pointer are
// exactly the LDS offset (ISA 10.2 aperture rule).
__device__ __forceinline__ void async_load_b128(const float* lds_dst,
                                                const float* gsrc) {
  const uint32_t lds_off = (uint32_t)(uintptr_t)lds_dst;
  asm volatile("global_load_async_to_lds_b128 %0, %1, off"
               :: "v"(lds_off), "v"(gsrc) : "memory");
}

__device__ __forceinline__ void wait_async0() {
#if __has_builtin(__builtin_amdgcn_s_wait_asynccnt)
  __builtin_amdgcn_s_wait_asynccnt(0);
#else
  asm volatile("s_wait_asynccnt 0x0" ::: "memory");
#endif
}

// ---------------------------------------------------------------------------
// Kernel 1: C[8192 x 1024] = X[8192 x 1024] @ W[1024 x 1024]
// Workgroup: 256 threads = 8 waves. WG tile 64(M) x 128(N).
// Wave w: M rows (w&3)*16, N cols (w>>2)*64 -> 4 accumulators of 16x16.
// Xs padded to 20 floats/row (bank-conflict-free frag reads, 16B-aligned
// b128 staging). W staged TRANSPOSED so each B fragment is one ds_load_b64.
// ---------------------------------------------------------------------------
__global__ __launch_bounds__(256) void qkv_gemm_kernel(
    const float* __restrict__ X,
    const float* __restrict__ Wq, const float* __restrict__ Wk,
    const float* __restrict__ Wv,
    float* __restrict__ Qo, float* __restrict__ Ko, float* __restrict__ Vo)
{
  const float* W; float* O;
  if (blockIdx.z == 0)      { W = Wq; O = Qo; }
  else if (blockIdx.z == 1) { W = Wk; O = Ko; }
  else                      { W = Wv; O = Vo; }

  __shared__ float Xs[64][20];     // 5 KB   (row stride 80B: 16B aligned)
  __shared__ float Wst[128][20];   // 10 KB  (transposed W tile, padded)

  const int tid  = threadIdx.x;
  const int lane = tid & 31;
  const int w    = tid >> 5;
  const int hl   = lane & 15;     // lane within half-wave
  const int hi   = lane >> 4;     // half-wave -> which K pair of the frag
  const int mrow = (w & 3) * 16;
  const int nseg = (w >> 2) * 64;
  const int gm0  = blockIdx.y * 64;
  const int gn0  = blockIdx.x * 128;

  v8f acc[4];
  for (int t = 0; t < 4; ++t) acc[t] = (v8f){};

  for (int k0 = 0; k0 < DIM; k0 += 16) {
    // Async-stage X tile 64x16 (one b128 DMA per thread, no VGPR round-trip)
    {
      const int row = tid >> 2;
      const int c4  = (tid & 3) << 2;
      async_load_b128(&Xs[row][c4],
                      X + (size_t)(gm0 + row) * DIM + k0 + c4);
    }
    // Stage W tile 16x128 transposed: Wst[col][k]
    for (int i = tid; i < 512; i += 256) {
      const int row = i >> 5;              // k within tile (0..15)
      const int c4  = (i & 31) << 2;       // col (0..124)
      const float4 wv =
          *(const float4*)(W + (size_t)(k0 + row) * DIM + gn0 + c4);
      Wst[c4 + 0][row] = wv.x;
      Wst[c4 + 1][row] = wv.y;
      Wst[c4 + 2][row] = wv.z;
      Wst[c4 + 3][row] = wv.w;
    }
    wait_async0();
    __syncthreads();

    for (int kk = 0; kk < 16; kk += 4) {
      const int ka = kk + hi * 2;          // K pair for this half-wave
      v2f a;                               // A[m][k] = X[m][k] (one b64)
      a.x = Xs[mrow + hl][ka];
      a.y = Xs[mrow + hl][ka + 1];
      for (int t = 0; t < 4; ++t) {
        v2f bf;                            // B[k][n] = W[k][n] (one b64)
        bf.x = Wst[nseg + t * 16 + hl][ka];
        bf.y = Wst[nseg + t * 16 + hl][ka + 1];
        acc[t] = wmma_f32(a, bf, acc[t]);
      }
    }
    __syncthreads();
  }

  // Store: C/D layout -> VGPR r holds M = r (lanes 0-15) / r+8 (lanes 16-31)
  for (int t = 0; t < 4; ++t)
    for (int r = 0; r < 8; ++r) {
      const int row = gm0 + mrow + r + hi * 8;
      const int col = gn0 + nseg + t * 16 + hl;
      O[(size_t)row * DIM + col] = acc[t][r];
    }
}

// ---------------------------------------------------------------------------
// Kernel 2: flash attention. Grid: (SEQ/16, BATCH). Block: 256 threads.
// Each workgroup handles one 16-row Q block of one batch.
// Wave w owns output columns [w*128, w*128+128).
// Ks padded to 516 floats/row: 516 mod 64 = 4, so 16-lane B-frag reads
// (stride 516) are conflict-free, and 516*4 = 2064B keeps b128 alignment.
// ---------------------------------------------------------------------------
__global__ __launch_bounds__(256) void flash_attn_kernel(
    const float* __restrict__ Q, const float* __restrict__ K,
    const float* __restrict__ V, float* __restrict__ Out)
{
  const int b  = blockIdx.y;
  const int q0 = blockIdx.x * 16;
  const float* Qb = Q + (size_t)b * SEQ * DIM;
  const float* Kb = K + (size_t)b * SEQ * DIM;
  const float* Vb = V + (size_t)b * SEQ * DIM;
  float*       Ob = Out + (size_t)b * SEQ * DIM;

  __shared__ float Ks[16][516];          // ~32 KB, K tile staged in 2 D-halves
  __shared__ float Spart[8][16][16];     // 8 KB, per-wave partial scores
  __shared__ float Ssum[16][16];         // 1 KB, reduced scores
  __shared__ float Pbuf[16][20];         // padded: conflict-free A-frag reads
  __shared__ float rowM[16], rowL[16], rowC[16];

  const int tid  = threadIdx.x;
  const int lane = tid & 31;
  const int w    = tid >> 5;
  const int hl   = lane & 15;
  const int hi   = lane >> 4;
  const int c0   = w * 128;              // this wave's output-column slice

  // Preload Q fragments for this wave's D columns: phase p covers
  // D in [p*512 + w*64, p*512 + w*64 + 64)  -> 16 k-steps of 4 per phase.
  v2f qf[32];
  for (int p = 0; p < 2; ++p) {
    const int dbase = p * 512 + w * 64;
    for (int s = 0; s < 16; ++s) {
      const int d = dbase + s * 4 + hi * 2;
      v2f a;
      a.x = Qb[(size_t)(q0 + hl) * DIM + d];
      a.y = Qb[(size_t)(q0 + hl) * DIM + d + 1];
      qf[p * 16 + s] = a;
    }
  }

  v8f oacc[8];
  for (int t = 0; t < 8; ++t) oacc[t] = (v8f){};
  if (tid < 16) { rowM[tid] = -1e30f; rowL[tid] = 0.0f; }
  __syncthreads();

  for (int j = 0; j <= q0; j += 16) {
    // ---- scores: S_partial = Q_slice @ K_slice^T ------------------------
    v8f sacc = (v8f){};
    Ssum[tid >> 4][tid & 15] = 0.0f;   // not read until after reduction write

    for (int p = 0; p < 2; ++p) {
      __syncthreads();
      // Async-stage K rows j..j+15, D cols [p*512, p*512+512): 8 b128/thread
      for (int i = tid; i < 2048; i += 256) {
        const int row = i >> 7;
        const int c4  = (i & 127) << 2;
        async_load_b128(&Ks[row][c4],
                        Kb + (size_t)(j + row) * DIM + p * 512 + c4);
      }
      wait_async0();
      __syncthreads();

      const int dloc = w * 64;
      for (int s = 0; s < 16; ++s) {
        const int d = dloc + s * 4 + hi * 2;
        v2f bf;                           // B[k][n] = K[j+n][D=k] (one b64)
        bf.x = Ks[hl][d];
        bf.y = Ks[hl][d + 1];
        sacc = wmma_f32(qf[p * 16 + s], bf, sacc);
      }
    }

    // ---- reduce 8 partial score tiles ----------------------------------
    for (int r = 0; r < 8; ++r) Spart[w][r + hi * 8][hl] = sacc[r];
    __syncthreads();
    {
      const int m = tid >> 4, n = tid & 15;
      float s = 0.0f;
      for (int k = 0; k < 8; ++k) s += Spart[k][m][n];
      s *= SCALE;
      if (j + n > q0 + m) s = -1e30f;    // causal mask (only bites when j==q0)
      Ssum[m][n] = s;
    }
    __syncthreads();

    // ---- streaming softmax (one thread per row) ------------------------
    if (tid < 16) {
      const int m = tid;
      float mx = rowM[m];
      float rmax = mx;
      for (int n = 0; n < 16; ++n) rmax = fmaxf(rmax, Ssum[m][n]);
      const float corr = __expf(mx - rmax);
      float psum = 0.0f;
      for (int n = 0; n < 16; ++n) {
        const float pv = __expf(Ssum[m][n] - rmax);
        Pbuf[m][n] = pv;
        psum += pv;
      }
      rowL[m] = rowL[m] * corr + psum;
      rowM[m] = rmax;
      rowC[m] = corr;
    }
    __syncthreads();

    // ---- rescale running O and accumulate P @ V ------------------------
    for (int r = 0; r < 8; ++r) {
      const float c = rowC[r + hi * 8];
      for (int t = 0; t < 8; ++t) oacc[t][r] *= c;
    }
    for (int s = 0; s < 4; ++s) {
      const int kk = s * 4 + hi * 2;
      v2f a;                              // A[m][k] = P[m][k] (one b64)
      a.x = Pbuf[hl][kk];
      a.y = Pbuf[hl][kk + 1];
      for (int t = 0; t < 8; ++t) {
        v2f bf;                           // B[k][n] = V[j+k][c0 + t*16 + n]
        bf.x = Vb[(size_t)(j + kk) * DIM + c0 + t * 16 + hl];
        bf.y = Vb[(size_t)(j + kk + 1) * DIM + c0 + t * 16 + hl];
        oacc[t] = wmma_f32(a, bf, oacc[t]);
      }
    }
    __syncthreads();
  }

  // ---- finalize: divide by row sums, round like the reference ----------
  for (int r = 0; r < 8; ++r) {
    const float linv = 1.0f / rowL[r + hi * 8];
    const int row = q0 + r + hi * 8;
    for (int t = 0; t < 8; ++t) {
      float v = oacc[t][r] * linv;
      v = rintf(v * 10000.0f) * 0.0001f;  // jnp.round(x*1e4)/1e4 (ties-to-even)
      Ob[(size_t)row * DIM + c0 + t * 16 + hl] = v;
    }
  }
}

// ---------------------------------------------------------------------------
extern "C" void kernel_launch(void* const* d_in, const int* in_sizes, int n_in,
                              void* d_out, int out_size, void* d_ws, size_t ws_size,
                              hipStream_t stream) {
  const float* X  = (const float*)d_in[0];   // embedded [4,2048,1024]
  const float* Wk = (const float*)d_in[1];
  const float* Wq = (const float*)d_in[2];
  const float* Wv = (const float*)d_in[3];
  float* Out = (float*)d_out;

  const size_t NT = (size_t)NROWS * DIM;     // 8,388,608 elements each
  float* Qd = (float*)d_ws;
  float* Kd = Qd + NT;
  float* Vd = Kd + NT;

  // Stage 1: Q/K/V projections (z selects which weight/output)
  dim3 g1(DIM / 128, NROWS / 64, 3);
  qkv_gemm_kernel<<<g1, 256, 0, stream>>>(X, Wq, Wk, Wv, Qd, Kd, Vd);

  // Stage 2: causal flash attention
  dim3 g2(SEQ / 16, BATCH, 1);
  flash_attn_kernel<<<g2, 256, 0, stream>>>(Qd, Kd, Vd, Out);
}